// PersistentMemory_82832739270872
// MI455X (gfx1250) — compile-verified
//
#include <hip/hip_runtime.h>
#include <math.h>
#include <stdint.h>

#define D 1024
#define NSLOT 16
#define BATCH 8192
#define DECAY 0.995f
#define SHARP 5.0f

typedef __attribute__((ext_vector_type(2))) float v2f;
typedef __attribute__((ext_vector_type(8))) float v8f;

// ---------------------------------------------------------------------------
// WMMA f32 GEMM:  C(M x D) = A(M x D) @ W(D x D) + bias
// grid: (D/128, M/128, 2)   z=0 -> (Wv,bv,value), z=1 -> (Wq,bq,query)
// block: 256 threads = 8 waves; each wave computes a 16(M) x 128(N) strip
// (8 x v8f accumulators). W tile (32K x 128N, 16KB) is double-buffered in
// LDS via global_load_async_to_lds_b128 (ASYNCcnt pipeline), shared by all
// 8 waves. A is a direct per-lane b64 load (L2-resident).
// Fragment layouts per CDNA5 ISA 7.12.2 (32-bit A 16x4, 32-bit C 16x16).
//
// LDS bank swizzle: within each 32-column quadrant, rows with (k&2)!=0 are
// rotated by 16 columns. The two half-waves of a fragment load (k-pairs
// {0,1} vs {2,3}) then hit disjoint 16-bank halves; the rotation is linear
// in the 32-column quadrant index, so the 4 cooperative async copies per
// thread keep a constant 128B stride on BOTH the global and LDS side
// (required since the async op adds one IOFFSET to both addresses).
// ---------------------------------------------------------------------------
__global__ __launch_bounds__(256)
void gemm_wmma_kernel(const float* __restrict__ A,
                      const float* __restrict__ Wv, const float* __restrict__ bv,
                      const float* __restrict__ Wq, const float* __restrict__ bq,
                      float* __restrict__ value, float* __restrict__ query) {
    const float* W    = (blockIdx.z == 0) ? Wv : Wq;
    const float* bias = (blockIdx.z == 0) ? bv : bq;
    float*       C    = (blockIdx.z == 0) ? value : query;

    const int t    = threadIdx.x;
    const int wave = t >> 5;
    const int lane = t & 31;
    const int lm   = lane & 15;                 // row/col within 16
    const int kb   = (lane < 16) ? 0 : 2;       // K sub-pair held by this lane half

    const int mBase = blockIdx.y * 128 + wave * 16;
    const int nBase = blockIdx.x * 128;

    __shared__ __align__(16) float wtile[2][32 * 128];   // 2 x 16KB

    // cooperative async-copy mapping: thread t owns row r = t>>3 (0..31),
    // 4-col group c4 = (t&7)*4, and copies quadrants q=0..3 (stride 128B).
    const int r  = t >> 3;              // 0..31
    const int c4 = (t & 7) << 2;        // 0,4,...,28
    const int sw = (r & 2) << 3;        // write swizzle shift: 0 or 16
    const int cw = (c4 + sw) & 31;      // swizzled column within quadrant

    const float*   gBase = W + (size_t)r * D + nBase + c4;     // + kc*32*D
    const unsigned lBase0 = (unsigned)(uintptr_t)&wtile[0][r * 128 + cw];
    const unsigned lBase1 = (unsigned)(uintptr_t)&wtile[1][r * 128 + cw];

    auto issue_chunk = [&](int kc, int buf) {
        const float* g = gBase + (size_t)(kc * 32) * D;
        unsigned l = buf ? lBase1 : lBase0;
        asm volatile("global_load_async_to_lds_b128 %0, %1, off"
                     :: "v"(l), "v"(g) : "memory");
        asm volatile("global_load_async_to_lds_b128 %0, %1, off offset:128"
                     :: "v"(l), "v"(g) : "memory");
        asm volatile("global_load_async_to_lds_b128 %0, %1, off offset:256"
                     :: "v"(l), "v"(g) : "memory");
        asm volatile("global_load_async_to_lds_b128 %0, %1, off offset:384"
                     :: "v"(l), "v"(g) : "memory");
    };

    const float* Arow = A + (size_t)(mBase + lm) * D;

    v8f acc[8];
    #pragma unroll
    for (int s = 0; s < 8; ++s) acc[s] = (v8f){};

    issue_chunk(0, 0);

    for (int kc = 0; kc < D / 32; ++kc) {
        const int buf = kc & 1;
        if (kc + 1 < D / 32) {
            issue_chunk(kc + 1, buf ^ 1);
            // oldest chunk's 4 async loads complete; newest 4 stay in flight
            asm volatile("s_wait_asynccnt 0x4" ::: "memory");
        } else {
            asm volatile("s_wait_asynccnt 0x0" ::: "memory");
        }
        __syncthreads();

        const float* Ar = Arow + kc * 32;
        #pragma unroll
        for (int ks = 0; ks < 8; ++ks) {
            const int kk = ks * 4 + kb;           // K row pair within chunk
            // A fragment (16x4 f32): lanes 0-15 K={0,1}, lanes 16-31 K={2,3}
            v2f a = *(const v2f*)(Ar + kk);

            const float* brow = &wtile[buf][kk * 128];
            const int shift = (kk & 2) << 3;      // read swizzle: 0 or 16
            #pragma unroll
            for (int s = 0; s < 8; ++s) {
                const int col  = s * 16 + lm;
                const int colp = (col & 96) | (((col & 31) + shift) & 31);
                v2f bf;
                bf.x = brow[colp];          // row kk
                bf.y = brow[128 + colp];    // row kk+1 (same swizzle: (kk+1)&2==kk&2)
                acc[s] = __builtin_amdgcn_wmma_f32_16x16x4_f32(
                    false, a, false, bf, (short)0, acc[s], false, false);
            }
        }
        __syncthreads();   // all waves done reading before buffer is refilled
    }

    // Store C: VGPR j holds row (mBase + j + (lane>=16 ? 8:0)), col = nBase + s*16 + lm
    const int rOff = (lane < 16) ? 0 : 8;
    float bsum[8];
    #pragma unroll
    for (int s = 0; s < 8; ++s) bsum[s] = bias[nBase + s * 16 + lm];
    #pragma unroll
    for (int j = 0; j < 8; ++j) {
        const size_t row = (size_t)(mBase + j + rOff);
        float* Crow = C + row * D + nBase + lm;
        #pragma unroll
        for (int s = 0; s < 8; ++s)
            Crow[s * 16] = acc[s][j] + bsum[s];
    }
}

// ---------------------------------------------------------------------------
// Fused per-row kernel: gate, addr-softmax, memory decay+write, attention read.
// One 256-thread block per batch row; each thread owns 4 strided elements.
// Single pass over the 512MB memory tensor (1 read + 1 write) -> HBM-bound.
// ---------------------------------------------------------------------------
__device__ __forceinline__ float wave_reduce(float v) {
    #pragma unroll
    for (int off = 16; off > 0; off >>= 1)
        v += __shfl_down(v, off, 32);
    return v;
}

__global__ __launch_bounds__(256)
void fused_update_kernel(const float* __restrict__ h,
                         const float* __restrict__ mem_in,
                         const float* __restrict__ Wg, const float* __restrict__ bg,
                         const float* __restrict__ Wa, const float* __restrict__ ba,
                         const float* __restrict__ value,
                         const float* __restrict__ query,
                         float* __restrict__ rv_out,
                         float* __restrict__ mem_out,
                         float* __restrict__ gate_out) {
    const int b = blockIdx.x;
    const int t = threadIdx.x;
    const int lane = t & 31;

    __shared__ float red[17];    // [0..15] addr logits, [16] gate dot
    __shared__ float sred[16];   // attention scores
    if (t < 17) red[t] = 0.0f;
    if (t < 16) sred[t] = 0.0f;
    __syncthreads();

    // ---- load h row (4 strided elements per thread, coalesced) ----
    float hreg[4];
    #pragma unroll
    for (int j = 0; j < 4; ++j) hreg[j] = h[(size_t)b * D + t + j * 256];

    // ---- gate + addr partial dots ----
    float gp = 0.0f;
    float ap[NSLOT];
    #pragma unroll
    for (int n = 0; n < NSLOT; ++n) ap[n] = 0.0f;
    #pragma unroll
    for (int j = 0; j < 4; ++j) {
        const int i = t + j * 256;
        const float hv = hreg[j];
        gp += hv * Wg[i];
        const float* war = Wa + (size_t)i * NSLOT;
        #pragma unroll
        for (int n = 0; n < NSLOT; ++n) ap[n] += hv * war[n];
    }
    {
        float rg = wave_reduce(gp);
        if (lane == 0) atomicAdd(&red[16], rg);
        #pragma unroll
        for (int n = 0; n < NSLOT; ++n) {
            float rn = wave_reduce(ap[n]);
            if (lane == 0) atomicAdd(&red[n], rn);
        }
    }
    __syncthreads();

    // ---- gate + addr softmax (computed redundantly by all threads) ----
    const float gate = 1.0f / (1.0f + __expf(-SHARP * (red[16] + bg[0])));
    float addr[NSLOT];
    float mx = -1e30f;
    #pragma unroll
    for (int n = 0; n < NSLOT; ++n) mx = fmaxf(mx, red[n] + ba[n]);
    float asum = 0.0f;
    #pragma unroll
    for (int n = 0; n < NSLOT; ++n) { addr[n] = __expf(red[n] + ba[n] - mx); asum += addr[n]; }
    const float ainv = 1.0f / asum;
    #pragma unroll
    for (int n = 0; n < NSLOT; ++n) addr[n] *= ainv;

    // ---- load value & query rows ----
    float vreg[4], qreg[4];
    #pragma unroll
    for (int j = 0; j < 4; ++j) {
        vreg[j] = value[(size_t)b * D + t + j * 256];
        qreg[j] = query[(size_t)b * D + t + j * 256];
    }

    // ---- single pass over memory: decay + gated write + score dot ----
    float mnew[NSLOT][4];
    #pragma unroll
    for (int n = 0; n < NSLOT; ++n) {
        const float ga = gate * addr[n];
        const size_t base = ((size_t)b * NSLOT + n) * D;
        float sp = 0.0f;
        #pragma unroll
        for (int j = 0; j < 4; ++j) {
            const int i = t + j * 256;
            float m = mem_in[base + i] * DECAY + ga * vreg[j];
            mnew[n][j] = m;
            mem_out[base + i] = m;
            sp += m * qreg[j];
        }
        float rs = wave_reduce(sp);
        if (lane == 0) atomicAdd(&sred[n], rs);
    }
    __syncthreads();

    // ---- attention softmax over 16 scores ----
    const float rscale = 0.03125f;   // 1/sqrt(1024)
    float smx = -1e30f;
    #pragma unroll
    for (int n = 0; n < NSLOT; ++n) smx = fmaxf(smx, sred[n] * rscale);
    float ssum = 0.0f;
    float attn[NSLOT];
    #pragma unroll
    for (int n = 0; n < NSLOT; ++n) { attn[n] = __expf(sred[n] * rscale - smx); ssum += attn[n]; }
    const float sinv = 1.0f / ssum;

    // ---- attention read ----
    #pragma unroll
    for (int j = 0; j < 4; ++j) {
        float acc = 0.0f;
        #pragma unroll
        for (int n = 0; n < NSLOT; ++n) acc += attn[n] * sinv * mnew[n][j];
        rv_out[(size_t)b * D + t + j * 256] = acc;
    }
    if (t == 0) gate_out[b] = gate;
}

// ---------------------------------------------------------------------------
extern "C" void kernel_launch(void* const* d_in, const int* in_sizes, int n_in,
                              void* d_out, int out_size, void* d_ws, size_t ws_size,
                              hipStream_t stream) {
    const float* h_t  = (const float*)d_in[0];
    const float* mem  = (const float*)d_in[1];
    const float* Wg   = (const float*)d_in[2];
    const float* bg   = (const float*)d_in[3];
    const float* Wa   = (const float*)d_in[4];
    const float* ba   = (const float*)d_in[5];
    const float* Wv   = (const float*)d_in[6];
    const float* bv   = (const float*)d_in[7];
    const float* Wq   = (const float*)d_in[8];
    const float* bq   = (const float*)d_in[9];

    float* out      = (float*)d_out;
    float* rv_out   = out;                                      // (B, D)
    float* mem_out  = out + (size_t)BATCH * D;                  // (B, NSLOT, D)
    float* gate_out = out + (size_t)BATCH * D + (size_t)BATCH * NSLOT * D;  // (B,)

    float* value = (float*)d_ws;                                // (B, D)
    float* query = value + (size_t)BATCH * D;                   // (B, D)

    // GEMMs: value = h@Wv+bv ; query = h@Wq+bq
    dim3 ggrid(D / 128, BATCH / 128, 2);
    gemm_wmma_kernel<<<ggrid, 256, 0, stream>>>(h_t, Wv, bv, Wq, bq, value, query);

    // Fused gate/addr/decay/write/read
    fused_update_kernel<<<BATCH, 256, 0, stream>>>(h_t, mem, Wg, bg, Wa, ba,
                                                   value, query,
                                                   rv_out, mem_out, gate_out);
}